// QoSGAE_77360950935940
// MI455X (gfx1250) — compile-verified
//
#include <hip/hip_runtime.h>
#include <hip/hip_bf16.h>

// -------- CDNA5 WMMA types: V_WMMA_F32_16X16X4_F32 (full fp32 matrix MAC) --------
typedef float v2f __attribute__((ext_vector_type(2)));
typedef float v8f __attribute__((ext_vector_type(8)));

#define NNODES 50000
#define NOBS   800000
#define NEDGE  (2 * NOBS)   // 1,600,000

// ---------------------------------------------------------------------------
// Zero a region (float4-granular; all our regions are multiples of 4 floats)
// ---------------------------------------------------------------------------
__global__ void zero_f32_kernel(float4* __restrict__ p, long n4) {
    long i = (long)blockIdx.x * blockDim.x + threadIdx.x;
    if (i < n4) p[i] = make_float4(0.f, 0.f, 0.f, 0.f);
}

// ---------------------------------------------------------------------------
// Layer-1 scatter: agg1[dst] += ew * x[src], feature dim 2 (thread per edge)
// ---------------------------------------------------------------------------
__global__ void scatter_l1_kernel(const float* __restrict__ x,
                                  const int* __restrict__ src,
                                  const int* __restrict__ dst,
                                  const float* __restrict__ ew,
                                  float* __restrict__ agg) {
    int e = blockIdx.x * blockDim.x + threadIdx.x;
    if (e >= NEDGE) return;
    int s = src[e], d = dst[e];
    float w = ew[e];
    float2 v = *(const float2*)(x + 2l * s);
    atomicAdd(&agg[2l * d + 0], w * v.x);
    atomicAdd(&agg[2l * d + 1], w * v.y);
}

// ---------------------------------------------------------------------------
// Layer-1 node update: h1 = relu(agg1 @ W1_rel + b1 + x @ W1_root), K=2, N=128
// (K=2 is too skinny for WMMA; 4 FMAs per output on the VALU is optimal)
// ---------------------------------------------------------------------------
__global__ void layer1_kernel(const float* __restrict__ agg, const float* __restrict__ x,
                              const float* __restrict__ Wrel, const float* __restrict__ b,
                              const float* __restrict__ Wroot, float* __restrict__ h) {
    int t = blockIdx.x * blockDim.x + threadIdx.x;
    if (t >= NNODES * 128) return;
    int n = t >> 7, j = t & 127;
    float a0 = agg[2 * n], a1 = agg[2 * n + 1];
    float x0 = x[2 * n],  x1 = x[2 * n + 1];
    float v = fmaf(a0, Wrel[j], fmaf(a1, Wrel[128 + j], b[j]));
    v = fmaf(x0, Wroot[j], fmaf(x1, Wroot[128 + j], v));
    h[t] = fmaxf(v, 0.f);
}

// ---------------------------------------------------------------------------
// Wave-per-edge scatter: agg[dst] += ew * h[src]. FPL floats per lane
// (F = 32*FPL). Gather is a contiguous 256B/512B burst per edge; atomics hit
// the 192MB L2 (tables are << L2), so no HBM round trips.
// ---------------------------------------------------------------------------
template <int FPL>
__global__ void scatter_wave_kernel(const float* __restrict__ h,
                                    const int* __restrict__ src,
                                    const int* __restrict__ dst,
                                    const float* __restrict__ ew,
                                    float* __restrict__ agg) {
    long tid = (long)blockIdx.x * blockDim.x + threadIdx.x;
    long e = tid >> 5;
    if (e >= NEDGE) return;
    int lane = (int)(tid & 31);
    int s = src[e], d = dst[e];
    float w = ew[e];
    const int F = FPL * 32;
    const float* hp = h + (long)s * F + lane * FPL;
    float* ap = agg + (long)d * F + lane * FPL;
    float v[FPL];
#pragma unroll
    for (int i = 0; i < FPL; ++i) v[i] = hp[i];
#pragma unroll
    for (int i = 0; i < FPL; ++i) atomicAdd(ap + i, w * v[i]);
}

// ---------------------------------------------------------------------------
// Fused GraphConv GEMM on fp32 WMMA:
//   out = act(A1 @ Wrel + A2 @ Wroot + bias),  A* : (NNODES,K), W* : (K,N)
// Block = 128 thr = 4 waves; each wave owns one 16x16 output tile.
// The 16-column weight slab for this blockIdx.y is staged in LDS (no bank
// conflicts: the two lane-halves read rows 2 apart -> 32 banks apart).
// A-fragment (ISA 7.12.2, 32-bit A 16x4): lane L holds A[m=L%16][kb..kb+1],
// kb = (L/16)*2. B mirrors with N striped across lanes.
// ---------------------------------------------------------------------------
template <int K, int N, bool RELU>
__global__ void conv_gemm_wmma_kernel(const float* __restrict__ A1, const float* __restrict__ Wrel,
                                      const float* __restrict__ A2, const float* __restrict__ Wroot,
                                      const float* __restrict__ bias, float* __restrict__ out) {
    __shared__ float lrel[K * 16];
    __shared__ float lroot[K * 16];
    const int nt = blockIdx.y;
    for (int i = threadIdx.x; i < K * 16; i += blockDim.x) {
        int k = i >> 4, nn = i & 15;
        lrel[i]  = Wrel [k * N + nt * 16 + nn];
        lroot[i] = Wroot[k * N + nt * 16 + nn];
    }
    __syncthreads();

    const int wave  = threadIdx.x >> 5;
    const int lane  = threadIdx.x & 31;
    const int mtile = blockIdx.x * 4 + wave;
    if (mtile * 16 >= NNODES) return;

    const int mrow = mtile * 16 + (lane & 15);
    const int kb   = (lane >> 4) * 2;
    const int nloc = lane & 15;

    v8f c = {};
#pragma unroll 4
    for (int k = 0; k < K; k += 4) {
        v2f a = *(const v2f*)(A1 + (long)mrow * K + k + kb);
        v2f bf;
        bf.x = lrel[(k + kb) * 16 + nloc];
        bf.y = lrel[(k + kb + 1) * 16 + nloc];
        c = __builtin_amdgcn_wmma_f32_16x16x4_f32(false, a, false, bf, (short)0, c, false, false);
    }
#pragma unroll 4
    for (int k = 0; k < K; k += 4) {
        v2f a = *(const v2f*)(A2 + (long)mrow * K + k + kb);
        v2f bf;
        bf.x = lroot[(k + kb) * 16 + nloc];
        bf.y = lroot[(k + kb + 1) * 16 + nloc];
        c = __builtin_amdgcn_wmma_f32_16x16x4_f32(false, a, false, bf, (short)0, c, false, false);
    }

    const int   n     = nt * 16 + nloc;
    const float bv    = bias[n];
    const int   mbase = mtile * 16 + (lane >> 4) * 8;   // C VGPR r -> row r + half*8
#pragma unroll
    for (int r = 0; r < 8; ++r) {
        float v = c[r] + bv;
        if (RELU) v = fmaxf(v, 0.f);
        out[(long)(mbase + r) * N + n] = v;
    }
}

// ---------------------------------------------------------------------------
// Decoder: pred[e] = relu([z[s]|z[d]] @ dec_W1 + b1) @ dec_W2 + b2
// One wave per 16 edges. A-fragment gathered straight from z (k<64 -> z[s],
// k>=64 -> z[d]; k-steps never straddle 64). Hidden 16x128 activation never
// hits memory: each 16x16 WMMA tile is relu'd and folded into a per-row dot
// with dec_W2 in registers, reduced with 16-lane shfl_xor at the end.
// ---------------------------------------------------------------------------
__global__ void decoder_wmma_kernel(const float* __restrict__ z,
                                    const int* __restrict__ src,
                                    const int* __restrict__ dst,
                                    const float* __restrict__ W1,
                                    const float* __restrict__ b1,
                                    const float* __restrict__ W2,
                                    const float* __restrict__ b2,
                                    float* __restrict__ pred) {
    const int  wave  = threadIdx.x >> 5;
    const int  lane  = threadIdx.x & 31;
    const long etile = (long)blockIdx.x * 8 + wave;
    const long e0    = etile * 16;
    const long e     = e0 + (lane & 15);
    const int  s = src[e], d = dst[e];
    const int  kb   = (lane >> 4) * 2;
    const int  nloc = lane & 15;
    const float* zs = z + (long)s * 64;
    const float* zd = z + (long)d * 64;

    float acc[8];
#pragma unroll
    for (int r = 0; r < 8; ++r) acc[r] = 0.f;

#pragma unroll 1
    for (int nt = 0; nt < 8; ++nt) {
        v8f c = {};
#pragma unroll 4
        for (int k = 0; k < 64; k += 4) {
            v2f a = *(const v2f*)(zs + k + kb);
            v2f bf;
            bf.x = W1[(k + kb) * 128 + nt * 16 + nloc];
            bf.y = W1[(k + kb + 1) * 128 + nt * 16 + nloc];
            c = __builtin_amdgcn_wmma_f32_16x16x4_f32(false, a, false, bf, (short)0, c, false, false);
        }
#pragma unroll 4
        for (int k = 0; k < 64; k += 4) {
            v2f a = *(const v2f*)(zd + k + kb);
            v2f bf;
            bf.x = W1[(64 + k + kb) * 128 + nt * 16 + nloc];
            bf.y = W1[(64 + k + kb + 1) * 128 + nt * 16 + nloc];
            c = __builtin_amdgcn_wmma_f32_16x16x4_f32(false, a, false, bf, (short)0, c, false, false);
        }
        const int   n  = nt * 16 + nloc;
        const float bb = b1[n], w2 = W2[n];
#pragma unroll
        for (int r = 0; r < 8; ++r)
            acc[r] += fmaxf(c[r] + bb, 0.f) * w2;
    }

    const float bout = b2[0];
#pragma unroll
    for (int r = 0; r < 8; ++r) {
        float v = acc[r];
        v += __shfl_xor(v, 1, 16);
        v += __shfl_xor(v, 2, 16);
        v += __shfl_xor(v, 4, 16);
        v += __shfl_xor(v, 8, 16);
        if (nloc == 0) {
            long row = e0 + (lane >> 4) * 8 + r;
            float p = v + bout;
            pred[row]        = p;   // pred_full = concat(pred, pred)
            pred[NOBS + row] = p;
        }
    }
}

// ---------------------------------------------------------------------------
extern "C" void kernel_launch(void* const* d_in, const int* in_sizes, int n_in,
                              void* d_out, int out_size, void* d_ws, size_t ws_size,
                              hipStream_t stream) {
    const float* x       = (const float*)d_in[0];
    const int*   ei      = (const int*)  d_in[1];
    const int*   src     = ei;
    const int*   dst     = ei + NEDGE;
    const float* ew      = (const float*)d_in[2];
    const float* W1_rel  = (const float*)d_in[4];
    const float* b1      = (const float*)d_in[5];
    const float* W1_root = (const float*)d_in[6];
    const float* W2_rel  = (const float*)d_in[7];
    const float* b2      = (const float*)d_in[8];
    const float* W2_root = (const float*)d_in[9];
    const float* W3_rel  = (const float*)d_in[10];
    const float* b3      = (const float*)d_in[11];
    const float* W3_root = (const float*)d_in[12];
    const float* dW1     = (const float*)d_in[13];
    const float* db1     = (const float*)d_in[14];
    const float* dW2     = (const float*)d_in[15];
    const float* db2     = (const float*)d_in[16];

    float* out  = (float*)d_out;
    float* pred = out;                  // 2*NOBS floats
    float* z    = out + 2l * NOBS;      // NNODES*64 floats

    // Workspace layout (floats): agg1 | agg2 | h1 | h2 ; agg3 aliases agg2
    // (agg2 is dead after the layer-2 GEMM). Total ~64.4 MB.
    float* ws   = (float*)d_ws;
    float* agg1 = ws;                                // 100,000
    float* agg2 = agg1 + 2l * NNODES;                // 6,400,000
    float* h1   = agg2 + 128l * NNODES;              // 6,400,000
    float* h2   = h1   + 128l * NNODES;              // 3,200,000
    float* agg3 = agg2;                              // reuse (3,200,000)

    const long z0 = (2l * NNODES + 128l * NNODES) / 4;   // agg1+agg2 in float4
    zero_f32_kernel<<<(int)((z0 + 255) / 256), 256, 0, stream>>>((float4*)agg1, z0);

    // Layer 1
    scatter_l1_kernel<<<(NEDGE + 255) / 256, 256, 0, stream>>>(x, src, dst, ew, agg1);
    layer1_kernel<<<(NNODES * 128 + 255) / 256, 256, 0, stream>>>(agg1, x, W1_rel, b1, W1_root, h1);

    // Layer 2: scatter 128-dim, then WMMA GEMM (M=50000,K=128,N=64)
    {
        long nthr = (long)NEDGE * 32;
        scatter_wave_kernel<4><<<(int)((nthr + 255) / 256), 256, 0, stream>>>(h1, src, dst, ew, agg2);
        dim3 g((3125 + 3) / 4, 64 / 16);   // 3125 m-tiles, 4 n-tiles
        conv_gemm_wmma_kernel<128, 64, true><<<g, 128, 0, stream>>>(agg2, W2_rel, h1, W2_root, b2, h2);
    }

    // Layer 3: re-zero aliased agg3, scatter 64-dim, WMMA GEMM -> z (no relu)
    {
        const long z1 = (64l * NNODES) / 4;
        zero_f32_kernel<<<(int)((z1 + 255) / 256), 256, 0, stream>>>((float4*)agg3, z1);
        long nthr = (long)NEDGE * 32;
        scatter_wave_kernel<2><<<(int)((nthr + 255) / 256), 256, 0, stream>>>(h2, src, dst, ew, agg3);
        dim3 g((3125 + 3) / 4, 64 / 16);
        conv_gemm_wmma_kernel<64, 64, false><<<g, 128, 0, stream>>>(agg3, W3_rel, h2, W3_root, b3, z);
    }

    // Decoder: 8 waves/block, 16 edges/wave -> 128 edges/block, 800000/128 = 6250 blocks
    decoder_wmma_kernel<<<NOBS / 128, 256, 0, stream>>>(z, src, dst, dW1, db1, dW2, db2, pred);
}